// dtw_loss_63015760167658
// MI455X (gfx1250) — compile-verified
//
#include <hip/hip_runtime.h>
#include <hip/hip_bf16.h>

#define TLEN 512
#define NPAIR 128
#define NTHR 128   // 4 rows per thread
#define NSTEP 255  // blocked anti-diagonals: s = t + cb, t,cb in [0,127]
#define BIGF 1e30f

#if defined(__has_builtin)
#if __has_builtin(__builtin_amdgcn_global_load_async_to_lds_b32)
#define HAVE_ASYNC_LDS 1
#endif
#if __has_builtin(__builtin_amdgcn_s_wait_asynccnt)
#define HAVE_WAIT_ASYNC 1
#endif
#if __has_builtin(__builtin_amdgcn_wmma_f32_16x16x4_f32)
#define HAVE_WMMA4 1
#endif
#endif

typedef __attribute__((address_space(1))) int as1_int;
typedef __attribute__((address_space(3))) int as3_int;

typedef __attribute__((ext_vector_type(2))) float v2f;
typedef __attribute__((ext_vector_type(8))) float v8f;

// One workgroup (4 waves) per (b,c) pair. Thread t owns DP rows 4t+1..4t+4.
// Blocked wavefront: at step s, thread t computes the 4x4 tile at column block
// cb = s - t (columns 4cb+1..4cb+4). Left column carried in registers; the
// tile's bottom row (5 floats: corner + 4 values) is handed to thread t+1
// through a parity-double-buffered LDS array. DP row 0 is synthesized as
// constants in thread 0. Per-cell arithmetic matches the reference exactly:
//   v = |x[i-1]-y[j-1]| + fmin(fmin(up, left), diag)
__global__ __launch_bounds__(NTHR) void dtw_tiled_kernel(
    const float* __restrict__ xin, const float* __restrict__ yin,
    float* __restrict__ partial) {
  __shared__ float ys[TLEN];
  __shared__ float H[2][NTHR][8];  // [parity][boundary row-block][5 used, pad 8]

  const int t = threadIdx.x;
  const int pair = blockIdx.x;
  const float* xp = xin + pair * TLEN;
  const float* yp = yin + pair * TLEN;

  // Stage y into LDS (CDNA5 async DMA path, ASYNCcnt-tracked).
#if HAVE_ASYNC_LDS
#pragma unroll
  for (int k = 0; k < 4; ++k) {
    __builtin_amdgcn_global_load_async_to_lds_b32(
        (as1_int*)(yp + 4 * t + k), (as3_int*)&ys[4 * t + k], 0, 0);
  }
#else
  *(float4*)&ys[4 * t] = *(const float4*)(yp + 4 * t);
#endif

  // x rows for this thread are loop-invariant: keep in registers.
  const float4 xv = *(const float4*)(xp + 4 * t);
  const float x4[4] = {xv.x, xv.y, xv.z, xv.w};

  // Left column entering column-block 0: dp[4t+1+r][0] = BIG.
  float L[4] = {BIGF, BIGF, BIGF, BIGF};

#if HAVE_ASYNC_LDS
#if HAVE_WAIT_ASYNC
  __builtin_amdgcn_s_wait_asynccnt(0);
#else
  asm volatile("s_wait_asynccnt 0" ::: "memory");
#endif
#endif
  __syncthreads();

  for (int s = 0; s < NSTEP; ++s) {
    const int cb = s - t;
    if (cb >= 0 && cb < NTHR) {
      // Top row dp[4t][4cb .. 4cb+4]: from thread t-1's hand-off, or the DP
      // boundary row for t==0 (dp[0][0]=0, dp[0][j>0]=BIG).
      float up0, up1, up2, up3, up4;
      if (t == 0) {
        up0 = (cb == 0) ? 0.0f : BIGF;
        up1 = BIGF; up2 = BIGF; up3 = BIGF; up4 = BIGF;
      } else {
        const int rp = (s + 1) & 1;  // parity written at step s-1
        const float4 h4 = *(const float4*)&H[rp][t][0];
        up0 = h4.x; up1 = h4.y; up2 = h4.z; up3 = h4.w;
        up4 = H[rp][t][4];
      }

      const float4 yv = *(const float4*)&ys[4 * cb];

#pragma unroll
      for (int r = 0; r < 4; ++r) {
        const float xr = x4[r];
        const float left0 = L[r];  // dp[4t+1+r][4cb]
        const float v1 = fabsf(xr - yv.x) + fminf(fminf(up1, left0), up0);
        const float v2 = fabsf(xr - yv.y) + fminf(fminf(up2, v1), up1);
        const float v3 = fabsf(xr - yv.z) + fminf(fminf(up3, v2), up2);
        const float v4 = fabsf(xr - yv.w) + fminf(fminf(up4, v3), up3);
        // Next row's top: {dp[i][4cb]=left0, v1..v4}
        up0 = left0; up1 = v1; up2 = v2; up3 = v3; up4 = v4;
        L[r] = v4;  // left column for the next column-block
      }

      // Hand bottom row dp[4t+4][4cb..4cb+4] = {up0..up4} to thread t+1.
      if (t < NTHR - 1) {
        const int wp = s & 1;
        float4 hv; hv.x = up0; hv.y = up1; hv.z = up2; hv.w = up3;
        *(float4*)&H[wp][t + 1][0] = hv;
        H[wp][t + 1][4] = up4;
      }
    }
    __syncthreads();
  }

  // Thread 127 finished tile (127,127): L[3] = dp[512][512].
  if (t == NTHR - 1) partial[pair] = L[3] * (1.0f / TLEN);
}

// Deterministic 128-way sum via two chained V_WMMA_F32_16X16X4_F32 with B = ones:
// D = A2*1 + (A1*1 + 0)  =>  every column n: D[m][n] = rowsum(m).
// Lane<16 VGPRs hold rows 0..7, lane>=16 hold rows 8..15; 8 local adds + shfl_xor(16).
__global__ __launch_bounds__(32) void dtw_reduce_kernel(
    const float* __restrict__ partial, float* __restrict__ out) {
  const int lane = threadIdx.x;  // exactly one wave32, EXEC all ones
#if HAVE_WMMA4
  const int m = lane & 15;
  const int k0 = (lane < 16) ? 0 : 2;  // A 16x4 layout: lanes 16-31 hold K=2,3
  const int base = 4 * m + k0;

  v2f a1, a2;
  a1.x = partial[base];
  a1.y = partial[base + 1];
  a2.x = partial[64 + base];
  a2.y = partial[64 + base + 1];
  v2f b = {1.0f, 1.0f};  // all-ones B: lane layout irrelevant
  v8f c = {};

  c = __builtin_amdgcn_wmma_f32_16x16x4_f32(false, a1, false, b, (short)0, c,
                                            false, false);
  c = __builtin_amdgcn_wmma_f32_16x16x4_f32(false, a2, false, b, (short)0, c,
                                            false, false);

  float s = ((c[0] + c[1]) + (c[2] + c[3])) + ((c[4] + c[5]) + (c[6] + c[7]));
  // lanes 0-15: sum(rows 0-7); lanes 16-31: sum(rows 8-15)
  float total = s + __shfl_xor(s, 16, 32);
  if (lane == 0) out[0] = total * 0.125f;  // mean over C=8, summed over B
#else
  if (lane == 0) {
    float s = 0.0f;
    for (int i = 0; i < NPAIR; ++i) s += partial[i];
    out[0] = s * 0.125f;
  }
#endif
}

extern "C" void kernel_launch(void* const* d_in, const int* in_sizes, int n_in,
                              void* d_out, int out_size, void* d_ws,
                              size_t ws_size, hipStream_t stream) {
  (void)in_sizes;
  (void)n_in;
  (void)out_size;
  (void)ws_size;
  const float* x = (const float*)d_in[0];  // inputs  (B,C,T) f32
  const float* y = (const float*)d_in[1];  // targets (B,C,T) f32
  float* partial = (float*)d_ws;           // 128 floats scratch
  float* out = (float*)d_out;              // 1 float

  dtw_tiled_kernel<<<NPAIR, NTHR, 0, stream>>>(x, y, partial);
  dtw_reduce_kernel<<<1, 32, 0, stream>>>(partial, out);
}